// LengthRegulator_67276367725261
// MI455X (gfx1250) — compile-verified
//
#include <hip/hip_runtime.h>

// ---------------- CDNA5 WMMA types ----------------
typedef float v2f __attribute__((ext_vector_type(2)));
typedef float v8f __attribute__((ext_vector_type(8)));

#define Bsz   16
#define Lsz   512
#define Dsz   256
#define Fsz   256
#define Msz   2048
#define KTOT  768      // C_in * 3 taps
#define KSTEP 32
#define NSTEP (KTOT / KSTEP)   // 24
#define ASTR  36       // 16 x KSTEP A tile, padded stride (bank-conflict free)
#define BSTR  36       // 256 x KSTEP B tile (stored [col][k]), padded stride
#define HSTR  260      // 16 x 256 epilogue tile, padded stride
#define TILEF (16 * ASTR + 256 * BSTR)   // floats per ping-pong buffer = 9792

// Wt[f*768 + s*256 + d] = conv_w[f, d, s]   (conv_w is (F, C, 3) row-major)
// One launch transposes both conv layers' weights (blocks [0,768) -> w1, [768,1536) -> w2).
__global__ void wt_transpose_kernel(const float* __restrict__ w1,
                                    const float* __restrict__ w2,
                                    float* __restrict__ wt1,
                                    float* __restrict__ wt2) {
    int bid = blockIdx.x;
    const float* w = w1;
    float* wt = wt1;
    if (bid >= 768) { bid -= 768; w = w2; wt = wt2; }
    int o = bid * 256 + threadIdx.x;             // 768*256 == 196608 exactly
    int f = o / KTOT;
    int rem = o - f * KTOT;
    int s = rem >> 8;
    int d = rem & 255;
    wt[o] = w[f * KTOT + d * 3 + s];
}

// One conv layer as fp32 WMMA GEMM (double-buffered LDS) + fused LayerNorm(+ReLU).
// mode 0: write (8192,256) activation rows.  mode 1: fuse 256->1 linear -> dup (8192,)
__global__ __launch_bounds__(512)
void conv_ln_kernel(const float* __restrict__ in,     // (B*L, 256)
                    const float* __restrict__ wt,     // (256, 768) transposed weights
                    const float* __restrict__ cbias,  // (256,)
                    const float* __restrict__ gamma,  // (256,)
                    const float* __restrict__ beta,   // (256,)
                    const float* __restrict__ linw,   // (256,)
                    const float* __restrict__ linb,   // (1,)
                    float* __restrict__ out,
                    int mode) {
    // two ping-pong staging buffers; epilogue tile aliases buffer 0
    __shared__ __align__(16) float smem[2 * TILEF];   // 78,336 B

    const int tid  = threadIdx.x;
    const int w    = tid >> 5;        // wave id 0..15 -> owns N-tile [w*16, w*16+16)
    const int lane = tid & 31;
    const int hi   = lane >> 4;       // K-half select for A/B fragments
    const int l15  = lane & 15;

    const int rb    = blockIdx.x;     // 512 row-blocks of 16 rows
    const int row0  = rb * 16;
    const int b     = row0 >> 9;      // 16-row tile never crosses a batch (16 | 512)
    const int lbase = row0 & 511;

    v8f acc = {0.f, 0.f, 0.f, 0.f, 0.f, 0.f, 0.f, 0.f};

    // A-fill: 16x32 = 512 elems, one per thread
    const int ar  = tid >> 5;
    const int at  = tid & 31;
    const int als = lbase + ar;
    // B-fill: 256x32 = 8192 elems, 16 per thread (four float4s of one half-row)
    const int bf = tid >> 1;
    const int bh = (tid & 1) * 16;

    float  aReg;
    float4 bReg0, bReg1, bReg2, bReg3;

    // ---- register prefetch of step `st` ----
    auto loadRegs = [&](int st) {
        const int kk0 = st * KSTEP;
        const int s   = kk0 >> 8;     // conv tap (KSTEP chunk never straddles taps)
        const int d0  = kk0 & 255;
        int ls = als + s - 1;
        aReg = 0.f;
        if (ls >= 0 && ls < Lsz)
            aReg = in[(b * Lsz + ls) * 256 + d0 + at];
        const float* src = wt + bf * KTOT + kk0 + bh;
        bReg0 = *(const float4*)(src + 0);
        bReg1 = *(const float4*)(src + 4);
        bReg2 = *(const float4*)(src + 8);
        bReg3 = *(const float4*)(src + 12);
    };
    auto storeTile = [&](float* buf) {
        buf[ar * ASTR + at] = aReg;                   // A tile [row][k]
        float* dst = buf + 16 * ASTR + bf * BSTR + bh; // B tile [col][k]
        *(float4*)(dst +  0) = bReg0;
        *(float4*)(dst +  4) = bReg1;
        *(float4*)(dst +  8) = bReg2;
        *(float4*)(dst + 12) = bReg3;
    };

    loadRegs(0);
    storeTile(smem);
    __syncthreads();

    int cur = 0;
    for (int st = 0; st < NSTEP; ++st) {
        if (st + 1 < NSTEP) loadRegs(st + 1);         // global loads in flight...

        const float* buf = smem + cur * TILEF;        // ...while WMMAs run off LDS
        const float* Ap = buf + l15 * ASTR + 2 * hi;
        const float* Bp = buf + 16 * ASTR + (w * 16 + l15) * BSTR + 2 * hi;
#pragma unroll
        for (int k4 = 0; k4 < KSTEP; k4 += 4) {
            v2f a  = *(const v2f*)(Ap + k4);
            v2f bb = *(const v2f*)(Bp + k4);
            acc = __builtin_amdgcn_wmma_f32_16x16x4_f32(
                false, a, false, bb, (short)0, acc, false, false);
        }
        if (st + 1 < NSTEP) {
            storeTile(smem + (cur ^ 1) * TILEF);      // other buffer: no race with readers
            __syncthreads();                          // single barrier per K-step
            cur ^= 1;
        }
    }
    __syncthreads();   // all waves done reading LDS before epilogue overwrites it

    // ---- epilogue: conv bias -> LDS tile -> per-row LayerNorm (+ReLU) ----
    float* Hs = smem;                                 // aliases buffer 0
    const int colw = w * 16 + l15;
    const float cb = cbias[colw];
#pragma unroll
    for (int r = 0; r < 8; ++r)
        Hs[(r + 8 * hi) * HSTR + colw] = acc[r] + cb; // C/D layout: M = r + 8*hi, N = l15
    __syncthreads();

    // wave w normalizes row w (256 values, 8 per lane)
    float vals[8];
    float sm = 0.f, sq = 0.f;
#pragma unroll
    for (int j = 0; j < 8; ++j) {
        float v = Hs[w * HSTR + lane + 32 * j];
        vals[j] = v; sm += v; sq += v * v;
    }
#pragma unroll
    for (int off = 16; off >= 1; off >>= 1) {
        sm += __shfl_xor(sm, off, 32);
        sq += __shfl_xor(sq, off, 32);
    }
    const float mu  = sm * (1.f / 256.f);
    float var = sq * (1.f / 256.f) - mu * mu;
    var = var < 0.f ? 0.f : var;
    const float inv = rsqrtf(var + 1e-5f);
    const int rowg  = row0 + w;

    if (mode == 0) {
#pragma unroll
        for (int j = 0; j < 8; ++j) {
            int col = lane + 32 * j;
            float hv = (vals[j] - mu) * inv * gamma[col] + beta[col];
            out[rowg * 256 + col] = hv > 0.f ? hv : 0.f;
        }
    } else {
        float dot = 0.f;
#pragma unroll
        for (int j = 0; j < 8; ++j) {
            int col = lane + 32 * j;
            float hv = (vals[j] - mu) * inv * gamma[col] + beta[col];
            hv = hv > 0.f ? hv : 0.f;
            dot += hv * linw[col];
        }
#pragma unroll
        for (int off = 16; off >= 1; off >>= 1)
            dot += __shfl_xor(dot, off, 32);
        if (lane == 0) {
            float dv = dot + linb[0];
            out[rowg] = dv > 0.f ? dv : 0.f;
        }
    }
}

// inclusive cumsum of durations per batch (Hillis-Steele in LDS)
__global__ void cumsum_kernel(const int* __restrict__ tgt, int* __restrict__ cs) {
    __shared__ int sbuf[512];
    const int bb = blockIdx.x, t = threadIdx.x;
    sbuf[t] = tgt[bb * 512 + t];
    __syncthreads();
    for (int off = 1; off < 512; off <<= 1) {
        int add = (t >= off) ? sbuf[t - off] : 0;
        __syncthreads();
        sbuf[t] += add;
        __syncthreads();
    }
    cs[bb * 512 + t] = sbuf[t];
}

// length regulation: 256-thread blocks, 4 output rows each; binary search cumsum
__global__ __launch_bounds__(256)
void regulate_kernel(const float* __restrict__ x,
                     const int* __restrict__ cs,
                     float* __restrict__ out) {
    const int r  = blockIdx.x * 4 + (threadIdx.x >> 6);  // row over B*M
    const int t  = threadIdx.x & 63;
    const int bb = r >> 11;                              // / 2048
    const int m  = r & 2047;
    const int* csb = cs + bb * 512;
    const int total = csb[511];
    float4 val = make_float4(0.f, 0.f, 0.f, 0.f);
    if (m < total) {
        int lo = 0, hiB = 512;                           // first l with csb[l] > m
        while (lo < hiB) {
            int mid = (lo + hiB) >> 1;
            if (csb[mid] <= m) lo = mid + 1; else hiB = mid;
        }
        val = ((const float4*)x)[(bb * 512 + lo) * 64 + t];
    }
    ((float4*)out)[(size_t)r * 64 + t] = val;
}

extern "C" void kernel_launch(void* const* d_in, const int* in_sizes, int n_in,
                              void* d_out, int out_size, void* d_ws, size_t ws_size,
                              hipStream_t stream) {
    (void)in_sizes; (void)n_in; (void)out_size; (void)ws_size;
    const float* x      = (const float*)d_in[0];
    const int*   target = (const int*)d_in[1];
    const float* w1  = (const float*)d_in[3];
    const float* cb1 = (const float*)d_in[4];
    const float* g1  = (const float*)d_in[5];
    const float* be1 = (const float*)d_in[6];
    const float* w2  = (const float*)d_in[7];
    const float* cb2 = (const float*)d_in[8];
    const float* g2  = (const float*)d_in[9];
    const float* be2 = (const float*)d_in[10];
    const float* lw  = (const float*)d_in[11];
    const float* lb  = (const float*)d_in[12];

    float* out = (float*)d_out;                      // (16,2048,256) then dup(16,512)
    float* dup = out + (size_t)Bsz * Msz * Dsz;      // +8,388,608
    // Scratch parked in the front of d_out; the regulator (last kernel) overwrites it.
    float* h1  = out;                                //  2,097,152 floats
    float* wt1 = out + 2097152;                      //    196,608 floats
    float* wt2 = out + 2097152 + 196608;             //    196,608 floats (ends < 2.5M)
    int*   cs  = (int*)d_ws;                         //     32 KB

    wt_transpose_kernel<<<1536, 256, 0, stream>>>(w1, w2, wt1, wt2);
    cumsum_kernel<<<Bsz, 512, 0, stream>>>(target, cs);

    conv_ln_kernel<<<512, 512, 0, stream>>>(x,  wt1, cb1, g1, be1, lw, lb, h1,  0);
    conv_ln_kernel<<<512, 512, 0, stream>>>(h1, wt2, cb2, g2, be2, lw, lb, dup, 1);

    regulate_kernel<<<Bsz * Msz / 4, 256, 0, stream>>>(x, cs, out);
}